// ScalarP1FunctionSpace_24232205484052
// MI455X (gfx1250) — compile-verified
//
#include <hip/hip_runtime.h>
#include <cstdint>

// P1 FE point evaluation on a fixed 17x17 uniform mesh (512 triangles).
// Direct O(1) cell lookup + per-triangle affine coefficients (8KB LUT),
// staged into LDS via the CDNA5 Tensor Data Mover. Memory-bound: ~25MB
// total traffic -> ~1.1us at 23.3 TB/s. WMMA is deliberately not used:
// the only matrix formulation is the 512x-redundant brute-force scan
// (~17 GFLOP of f32 WMMA), ~100x slower than this streaming kernel.

typedef unsigned int v4u __attribute__((ext_vector_type(4)));
typedef int          v8i __attribute__((ext_vector_type(8)));
typedef int          v4i __attribute__((ext_vector_type(4)));

#define N_TRI 512

// ---------------------------------------------------------------------------
// Kernel 1: fold weight[dofs], Minv, tri_A into per-triangle affine coeffs:
//   val(x,y) = w0 + w1*x + w2*y   for points inside triangle t.
// Derivation: st = (p - A) @ Minv ; val = c0 + s*(c1-c0) + t*(c2-c0)
//   w1 = m00*(c1-c0) + m01*(c2-c0)
//   w2 = m10*(c1-c0) + m11*(c2-c0)
//   w0 = c0 - Ax*w1 - Ay*w2
// ---------------------------------------------------------------------------
__global__ __launch_bounds__(256) void p1_prep_coeffs(
    const float* __restrict__ weight,
    const float* __restrict__ tri_A,
    const float* __restrict__ Minv,
    const int*   __restrict__ dofs,
    float4*      __restrict__ coeffs)
{
    int t = blockIdx.x * blockDim.x + threadIdx.x;
    if (t >= N_TRI) return;

    float Ax = tri_A[2 * t + 0];
    float Ay = tri_A[2 * t + 1];
    float m00 = Minv[4 * t + 0];   // Minv[t][0][0]
    float m01 = Minv[4 * t + 1];   // Minv[t][0][1]
    float m10 = Minv[4 * t + 2];   // Minv[t][1][0]
    float m11 = Minv[4 * t + 3];   // Minv[t][1][1]
    int d0 = dofs[3 * t + 0];
    int d1 = dofs[3 * t + 1];
    int d2 = dofs[3 * t + 2];
    float c0 = weight[d0];
    float e1 = weight[d1] - c0;
    float e2 = weight[d2] - c0;

    float w1 = fmaf(m00, e1, m01 * e2);
    float w2 = fmaf(m10, e1, m11 * e2);
    float w0 = c0 - fmaf(Ax, w1, Ay * w2);

    coeffs[t] = make_float4(w0, w1, w2, 0.0f);
}

// ---------------------------------------------------------------------------
// Point evaluation: O(1) structured-mesh lookup.
// Cell (ix,iy) = floor(16*p); triangle order in the reference scan is
// lower triangles (ly<lx) first [ids 0..255], upper (ly>=lx) last
// [ids 256..511] -> upper wins TOL-band ties, preserved here.
// ---------------------------------------------------------------------------
__device__ __forceinline__ float evalpt(float px, float py,
                                        const float4* __restrict__ c)
{
    float fx = px * 16.0f;
    float fy = py * 16.0f;
    int ix = (int)fx; ix = (ix > 15) ? 15 : ix;
    int iy = (int)fy; iy = (iy > 15) ? 15 : iy;
    float lx = fx - (float)ix;
    float ly = fy - (float)iy;
    // upper triangle's inside test is t = (ly-lx) > -TOL; it is scanned last
    int tri = (ix << 4) + iy + ((ly - lx > -1e-10f) ? 256 : 0);
    float4 w = c[tri];
    return fmaf(w.y, px, fmaf(w.z, py, w.x));
}

// ---------------------------------------------------------------------------
// Kernel 2: stream points. 8KB coefficient LUT staged into LDS by the
// Tensor Data Mover (one descriptor per workgroup, issued by wave 0).
// ---------------------------------------------------------------------------
__global__ __launch_bounds__(256) void p1_eval(
    const float4* __restrict__ x2,      // 2 points per float4
    const float4* __restrict__ coeffs,  // 512 x float4 in d_ws
    float2*       __restrict__ out2,    // 2 results per float2
    int npair)
{
    __shared__ float4 lds_c[N_TRI];     // 8192 bytes

#if __has_builtin(__builtin_amdgcn_tensor_load_to_lds) && \
    __has_builtin(__builtin_amdgcn_s_wait_tensorcnt)
    if (threadIdx.x == 0) {
        // ---- Tensor DMA Descriptor (D#), CDNA5 ISA ----
        uint32_t lds_addr = (uint32_t)(uintptr_t)(void*)&lds_c[0]; // low 32b of
        uint64_t ga       = (uint64_t)(uintptr_t)coeffs;           // flat = LDS off
        const uint32_t td0   = 2048u;   // tensor_dim0: 2048 dwords (8KB)
        const uint32_t tile0 = 2048u;   // tile_dim0 == tensor_dim0: 1-D copy

        v4u g0;
        g0.x = 1u;                                        // count=1 (valid), no gather
        g0.y = lds_addr;                                  // lds_addr [63:32]
        g0.z = (uint32_t)ga;                              // global_addr [95:64]
        g0.w = (uint32_t)((ga >> 32) & 0x1FFFFFFull)      // global_addr [120:96]
             | (2u << 30);                                // type=2 ("image")

        v8i g1;
        g1[0] = (int)(2u << 16);                          // data_size=4B, wg_mask=0
        g1[1] = (int)((td0 & 0xFFFFu) << 16);             // tensor_dim0[15:0]
        g1[2] = (int)(((td0 >> 16) & 0xFFFFu)             // tensor_dim0[31:16]
             | (1u << 16));                               // tensor_dim1 = 1
        g1[3] = (int)((tile0 & 0xFFFFu) << 16);           // tensor_dim1 hi | tile_dim0
        g1[4] = 0;                                        // tile_dim1=0, tile_dim2=0
        g1[5] = (int)td0;                                 // tensor_dim0_stride lo
        g1[6] = 0;
        g1[7] = 0;

        v4i g2 = {0, 0, 0, 0};                            // dims >1 unused
        v4i g3 = {0, 0, 0, 0};
        v8i g4 = {0, 0, 0, 0, 0, 0, 0, 0};                // 6-arg form extension grp

        __builtin_amdgcn_tensor_load_to_lds(g0, g1, g2, g3, g4, 0);
        __builtin_amdgcn_s_wait_tensorcnt(0);             // s_wait_tensorcnt 0
    }
#else
    for (int i = threadIdx.x; i < N_TRI; i += blockDim.x) lds_c[i] = coeffs[i];
#endif
    __syncthreads();

    const int stride = gridDim.x * blockDim.x;
    for (int i = blockIdx.x * blockDim.x + threadIdx.x; i < npair; i += stride) {
        float4 p = x2[i];                 // global_load_b128: two (x,y) points
        float2 r;
        r.x = evalpt(p.x, p.y, lds_c);    // ds_load_b128 gather + 4 FMA
        r.y = evalpt(p.z, p.w, lds_c);
        out2[i] = r;                      // global_store_b64
    }
}

// ---------------------------------------------------------------------------
// Host-side launch. Inputs (setup_inputs order):
//   0: x (8,262144,2) f32   1: weight (289) f32   2: tri_A (512,2) f32
//   3: Minv (512,2,2) f32   4: bbox (512,4) f32 (unused)  5: dofs (512,3) i32
// ---------------------------------------------------------------------------
extern "C" void kernel_launch(void* const* d_in, const int* in_sizes, int n_in,
                              void* d_out, int out_size, void* d_ws, size_t ws_size,
                              hipStream_t stream)
{
    const float* x      = (const float*)d_in[0];
    const float* weight = (const float*)d_in[1];
    const float* tri_A  = (const float*)d_in[2];
    const float* Minv   = (const float*)d_in[3];
    const int*   dofs   = (const int*)d_in[5];

    float4* coeffs = (float4*)d_ws;       // 512 * 16B = 8KB scratch

    p1_prep_coeffs<<<2, 256, 0, stream>>>(weight, tri_A, Minv, dofs, coeffs);

    int npair = in_sizes[0] / 4;          // float4 = 2 points
    p1_eval<<<2048, 256, 0, stream>>>((const float4*)x, coeffs,
                                      (float2*)d_out, npair);
}